// MyMaxUnpooling_64089501991261
// MI455X (gfx1250) — compile-verified
//
#include <hip/hip_runtime.h>

// SegNet max-unpooling, fused zero+scatter.
// x:   (B,H,W,C)  = (16,128,128,64) float32
// idx: (B,H,W,C)  int32 global flat offsets into (B,H2,W2,C)
// out: (B,H2,W2,C)= (16,256,256,64) float32
//
// Each thread owns one (b,h,w) window x 4 consecutive channels:
//   - 1x global_load_b128 of x, 1x global_load_b128 of idx (coalesced, NT)
//   - 4x global_store_b128 covering all 16 output elements it owns
//     (value where idx matches that slot, 0 elsewhere) -> no memset pass,
//     no atomics, exact single-writer coverage of the output.

typedef __attribute__((ext_vector_type(4))) float float4v;
typedef __attribute__((ext_vector_type(4))) int   int4v;

namespace {
constexpr int kC  = 64;          // channels
constexpr int kW  = 128;         // pooled width
constexpr int kH  = 128;         // pooled height
constexpr int kW2 = 256;         // unpooled width
constexpr int kH2 = 256;         // unpooled height
constexpr int kCV = kC / 4;      // float4 chunks along C (=16)
constexpr int kRow = kW2 * kC;   // floats between output rows (=16384)
}

__global__ __launch_bounds__(256) void max_unpool_fused_kernel(
    const float* __restrict__ x,
    const int*   __restrict__ idx,
    float*       __restrict__ out,
    int nvec)                       // number of float4 chunks in x
{
  const int tid = blockIdx.x * blockDim.x + threadIdx.x;
  if (tid >= nvec) return;

  // Streaming 128-bit loads, non-temporal (read exactly once).
  const float4v xv = __builtin_nontemporal_load(((const float4v*)x)  + tid);
  const int4v   iv = __builtin_nontemporal_load(((const int4v*)idx) + tid);

  // Decompose tid -> (b, h, w, c4). All pow2 -> shifts/masks.
  const int c4 = tid & (kCV - 1);
  int t        = tid >> 4;          // / kCV
  const int w  = t & (kW - 1);
  t          >>= 7;                 // / kW
  const int h  = t & (kH - 1);
  const int b  = t >> 7;            // / kH

  // Output offset of channel 4*c4 at the top-left of this 2x2 window.
  // Fits in int32: max ~67M < 2^31.
  const int base = ((b * kH2 + 2 * h) * kW2 + 2 * w) * kC + 4 * c4;

  float4v d00 = {}, d01 = {}, d10 = {}, d11 = {};
#pragma unroll
  for (int j = 0; j < 4; ++j) {
    const int   k = base + j;       // candidate slot (dr=0,dc=0) for channel 4*c4+j
    const int   m = iv[j];          // argmax flat index for this channel
    const float v = xv[j];
    d00[j] = (m == k)               ? v : 0.0f;   // (dr=0,dc=0)
    d01[j] = (m == k + kC)          ? v : 0.0f;   // (dr=0,dc=1)
    d10[j] = (m == k + kRow)        ? v : 0.0f;   // (dr=1,dc=0)
    d11[j] = (m == k + kRow + kC)   ? v : 0.0f;   // (dr=1,dc=1)
  }

  // Four 16B-aligned 128-bit streaming stores; per wave each is a
  // pair of contiguous 256B runs 256B apart (full cachelines), NT to bypass L2 (output
  // is 256 MB write-once > 192 MB L2).
  __builtin_nontemporal_store(d00, (float4v*)(out + base));
  __builtin_nontemporal_store(d01, (float4v*)(out + base + kC));
  __builtin_nontemporal_store(d10, (float4v*)(out + base + kRow));
  __builtin_nontemporal_store(d11, (float4v*)(out + base + kRow + kC));
}

extern "C" void kernel_launch(void* const* d_in, const int* in_sizes, int n_in,
                              void* d_out, int out_size, void* d_ws, size_t ws_size,
                              hipStream_t stream) {
  const float* x  = (const float*)d_in[0];
  const int*  idx = (const int*)d_in[1];
  float*      out = (float*)d_out;

  const int n    = in_sizes[0];       // 16*128*128*64 = 16,777,216
  const int nvec = n / 4;             // 4,194,304 float4 chunks
  const int block = 256;
  const int grid  = (nvec + block - 1) / block;  // 16384

  max_unpool_fused_kernel<<<grid, block, 0, stream>>>(x, idx, out, nvec);
}